// ProjectRenderScoreModel_23922967839382
// MI455X (gfx1250) — compile-verified
//
#include <hip/hip_runtime.h>

#define HH 200      // image size
#define HP 208      // padded to 13 * 16
#define NPTS 64     // cloud points
#define BSZ 8
#define NT 13       // HP / 16
#define MASK_ELEMS (BSZ * 3 * HH * HH)   // 960000 floats

typedef __attribute__((ext_vector_type(2))) float v2f;
typedef __attribute__((ext_vector_type(8))) float v8f;

// d_out scores region must start at zero (harness poisons d_out).
__global__ void prs_zero_scores(float* __restrict__ out) {
  int i = blockIdx.x * blockDim.x + threadIdx.x;
  if (i < BSZ * NPTS) out[MASK_ELEMS + i] = 0.0f;
}

__launch_bounds__(256)
__global__ void prs_main(const float* __restrict__ cam,    // (8,3,2,3)
                         const float* __restrict__ cloud,  // (8,64,3)
                         const float* __restrict__ mtgt,   // (8,3,200,200)
                         const float* __restrict__ sig,    // (8,64)
                         const float* __restrict__ p3b,    // (8,3)
                         const float* __restrict__ p2b,    // (8,3,2)
                         float* __restrict__ out,          // masks(8,3,200,200) ++ scores(8,64)
                         float* __restrict__ ws) {         // 24*200*200 scratch
  extern __shared__ float smem[];
  float* sT      = smem;                 // HH*HP  staged masks_target (pad cols zeroed)
  float* sEy     = sT   + HH * HP;       // NPTS*HP   Ey[n][y]
  float* sEx     = sEy  + NPTS * HP;     // NPTS*HP   Ex[n][x]
  float* sPtx    = sEx  + NPTS * HP;     // 64
  float* sPty    = sPtx + NPTS;          // 64
  float* sInv    = sPty + NPTS;          // 64   1/(2 sigma^2)
  float* sSy     = sInv + NPTS;          // 64   sum_y Ey[n][y]
  float* sSumx   = sSy  + NPTS;          // HP   per-column normalizer
  float* sScore  = sSumx + HP;           // 64
  float* sMaskSum = sScore + NPTS;       // 1

  const int tid = threadIdx.x;
  const int bc  = blockIdx.x;            // (b*3 + cam)
  const int b   = bc / 3;
  const float* T = mtgt + bc * HH * HH;

  // ---- Kick off async global->LDS DMA of masks_target tile FIRST, so it
  //      overlaps phase 0 (exp tables) and phase 1 (masks GEMM). 200 rows of
  //      50 x b128 chunks each; LDS rows padded to 208 floats (832B, 16B-aligned).
  for (int c = tid; c < HH * 50; c += 256) {
    int y = c / 50, j = c - y * 50;
    unsigned long long ga = (unsigned long long)(const void*)(T + y * HH + j * 4);
    unsigned ldsoff = (unsigned)(size_t)(void*)(sT + y * HP + j * 4);
    asm volatile("global_load_async_to_lds_b128 %0, %1, off"
                 :: "v"(ldsoff), "v"(ga) : "memory");
  }
  // zero the 8 pad columns of each staged row (lets phase 2 drop its guard)
  for (int i = tid; i < HH * 8; i += 256) {
    int y = i >> 3, q = i & 7;
    sT[y * HP + 200 + q] = 0.0f;
  }

  // ---- Phase 0: project points, build separable factor tables ----
  if (tid == 0) sMaskSum[0] = 0.0f;
  if (tid < NPTS) {
    sScore[tid] = 0.0f;
    float x0 = cloud[(b * NPTS + tid) * 3 + 0] + p3b[b * 3 + 0];
    float x1 = cloud[(b * NPTS + tid) * 3 + 1] + p3b[b * 3 + 1];
    float x2 = cloud[(b * NPTS + tid) * 3 + 2] + p3b[b * 3 + 2];
    const float* cc = cam + bc * 6;      // (2,3) affine for this (b,cam)
    float px = cc[0] * x0 + cc[1] * x1 + cc[2] * x2;
    float py = cc[3] * x0 + cc[4] * x1 + cc[5] * x2;
    // ref: p2d = p2 - p2base + 100 ; pts = clip((p2d-100)/100, -1, 1)
    float ptx = (px - p2b[bc * 2 + 0]) * 0.01f;
    float pty = (py - p2b[bc * 2 + 1]) * 0.01f;
    sPtx[tid] = fminf(fmaxf(ptx, -1.0f), 1.0f);
    sPty[tid] = fminf(fmaxf(pty, -1.0f), 1.0f);
    float s = sig[b * NPTS + tid];
    sInv[tid] = 1.0f / (2.0f * s * s);
  }
  __syncthreads();

  for (int idx = tid; idx < NPTS * HP; idx += 256) {
    int n = idx / HP, q = idx - n * HP;
    float ex = 0.0f, ey = 0.0f;
    if (q < HH) {
      float g  = -1.0f + (2.0f / 199.0f) * (float)q;   // linspace(-1,1,200)
      float dx = g - sPtx[n];
      float dy = g - sPty[n];
      ex = __expf(-dx * dx * sInv[n]);
      ey = __expf(-dy * dy * sInv[n]);
    }
    sEx[idx] = ex;
    sEy[idx] = ey;
  }
  __syncthreads();

  if (tid < NPTS) {
    float s = 0.0f;
    for (int y = 0; y < HH; ++y) s += sEy[tid * HP + y];
    sSy[tid] = s;
  }
  __syncthreads();
  if (tid < HP) {   // sum over (n, y) per column x  — the reference's axis-(2,3) normalizer
    float s = 0.0f;
    for (int n = 0; n < NPTS; ++n) s += sSy[n] * sEx[n * HP + tid];
    sSumx[tid] = fmaxf(s, 1e-8f);
  }
  __syncthreads();

  const int lane = tid & 31;
  const int wv   = tid >> 5;             // 8 waves
  const int row  = lane & 15;
  const int kb   = (lane >> 4) * 2;      // K sub-offset per A/B f32 16x16x4 layout
  float* wsMask  = ws + bc * HH * HH;

  // ---- Phase 1: masks[y,x] = min(sum_n Ey[n,y]*Ex[n,x], 1)  (208x208x64 GEMM) ----
  float lsum = 0.0f;
  for (int t = wv; t < NT * NT; t += 8) {
    int mt = t / NT, nt = t - mt * NT;
    int mBase = mt * 16 + row;           // y index (A rows)
    int nBase = nt * 16 + row;           // x index (B cols)
    v8f acc = 0.0f;
    for (int k0 = 0; k0 < NPTS; k0 += 4) {
      v2f a, bb;
      a.x  = sEy[(k0 + kb)     * HP + mBase];
      a.y  = sEy[(k0 + kb + 1) * HP + mBase];
      bb.x = sEx[(k0 + kb)     * HP + nBase];
      bb.y = sEx[(k0 + kb + 1) * HP + nBase];
      acc = __builtin_amdgcn_wmma_f32_16x16x4_f32(false, a, false, bb,
                                                  (short)0, acc, false, false);
    }
    int x     = nt * 16 + (lane & 15);
    int ybase = mt * 16 + ((lane < 16) ? 0 : 8);
#pragma unroll
    for (int i = 0; i < 8; ++i) {
      int y = ybase + i;
      if (x < HH && y < HH) {
        float v = fminf(acc[i], 1.0f);
        wsMask[y * HH + x] = v;
        lsum += v;
      }
    }
  }
  atomicAdd(sMaskSum, lsum);             // ds_add_f32
  __syncthreads();

  float invTotal = 1.0f / fmaxf(sMaskSum[0], 1e-8f);
  float* outMask = out + bc * HH * HH;
  for (int i = tid; i < HH * HH; i += 256)
    outMask[i] = wsMask[i] * invTotal;

  // ---- Drain the async DMA, then barrier so every wave sees the staged T ----
  asm volatile("s_wait_asynccnt 0" ::: "memory");
  __syncthreads();

  // ---- Phase 2: G[n,x] = sum_y Ey[n,y]*T[y,x]  (64x208x200 GEMM), then scores ----
  for (int t = wv; t < 4 * NT; t += 8) {
    int mt = t / NT, nt = t - mt * NT;
    int mBase = mt * 16 + row;           // point index (A rows, always < 64)
    int xcol  = nt * 16 + row;           // x (B cols; pad cols read staged zeros)
    v8f acc = 0.0f;
    for (int k0 = 0; k0 < HH; k0 += 4) {
      v2f a, bb;
      a.x  = sEy[mBase * HP + k0 + kb];
      a.y  = sEy[mBase * HP + k0 + kb + 1];
      bb.x = sT[(k0 + kb)     * HP + xcol];
      bb.y = sT[(k0 + kb + 1) * HP + xcol];
      acc = __builtin_amdgcn_wmma_f32_16x16x4_f32(false, a, false, bb,
                                                  (short)0, acc, false, false);
    }
    int x = nt * 16 + (lane & 15);
    if (x < HH) {
      float invs = 1.0f / sSumx[x];
      int nb = mt * 16 + ((lane < 16) ? 0 : 8);
#pragma unroll
      for (int i = 0; i < 8; ++i) {
        int n = nb + i;
        atomicAdd(&sScore[n], acc[i] * sEx[n * HP + x] * invs);
      }
    }
  }
  __syncthreads();

  if (tid < NPTS)
    atomicAdd(&out[MASK_ELEMS + b * NPTS + tid], sScore[tid] * (1.0f / 3.0f));
}

extern "C" void kernel_launch(void* const* d_in, const int* in_sizes, int n_in,
                              void* d_out, int out_size, void* d_ws, size_t ws_size,
                              hipStream_t stream) {
  const float* cam   = (const float*)d_in[0];   // cam_coeffs       (8,3,2,3)
  const float* cloud = (const float*)d_in[1];   // cloud_points     (8,64,3)
  const float* mtgt  = (const float*)d_in[2];   // masks_target     (8,3,200,200)
  const float* sig   = (const float*)d_in[3];   // blur_sigmas      (8,64)
  const float* p3b   = (const float*)d_in[4];   // points_3d_base   (8,3)
  const float* p2b   = (const float*)d_in[5];   // points_2d_base   (8,3,2)
  float* out = (float*)d_out;
  float* ws  = (float*)d_ws;                    // 24*200*200 f32 = 3.84 MB

  prs_zero_scores<<<1, 512, 0, stream>>>(out);

  size_t smem = (size_t)(HH * HP + 2 * NPTS * HP + 4 * NPTS + HP + NPTS + 1)
                * sizeof(float);                // ~275 KB of the 320 KB WGP LDS
  prs_main<<<BSZ * 3, 256, smem, stream>>>(cam, cloud, mtgt, sig, p3b, p2b, out, ws);
}